// Voxelization_2164663517790
// MI455X (gfx1250) — compile-verified
//
#include <hip/hip_runtime.h>

// Semantic Gaussian voxelization for MI455X (gfx1250).
// B=4 batches, 6890 surface verts each, splat into 128^3 grid, 7^3 stencil.
// Accumulators (~134 MB) fit in the 192 MB L2 -> scatter atomics resolve in L2;
// HBM traffic ~370 MB -> bandwidth-bound. Separable exp: 21 exps/vertex.
// Chunk staging uses the Tensor Data Mover (tensor_load_to_lds + s_wait_tensorcnt).

#define VRES 128
#define NSURF 6890
#define NVERT 8000
#define NBATCH 4
#define RADIUS 3
#define DIAM 7
#define NCELLS 343          // 7^3
#define CHUNK 32            // vertices staged per workgroup
#define INV_2SIG2 800.0f    // 1 / (2 * 0.025^2)
#define WEIGHT_EPS 1e-3f

#if defined(__gfx1250__) && __has_builtin(__builtin_amdgcn_tensor_load_to_lds)
#define USE_TDM 1
typedef __attribute__((ext_vector_type(4))) unsigned int v4u;
typedef __attribute__((ext_vector_type(8))) int          v8i;
typedef __attribute__((ext_vector_type(4))) int          v4i;

// Issue one TDM descriptor copying `nelem` contiguous dwords from global to LDS.
// D# layout per CDNA5 ISA ch.8: group0 = {flags, lds_addr, gaddr_lo, gaddr_hi|type},
// group1 = {data_size=4B, tensor_dim0=nelem, tensor_dim1=1, tile_dim0=nelem,
//           tile_dim1=1, stride0=nelem}.  Groups 2/3 unused (tile_dim2=0), extra
// operand group zero-filled (this toolchain's 6-arg builtin form).
__device__ __forceinline__ void tdm_load_1d(unsigned lds_addr, const void* gptr,
                                            unsigned nelem) {
  unsigned long long ga = (unsigned long long)(size_t)gptr;
  v4u g0;
  g0.x = 1u;                                              // count=1, user mode
  g0.y = lds_addr;                                        // LDS byte address
  g0.z = (unsigned)ga;                                    // global_addr[31:0]
  g0.w = (unsigned)((ga >> 32) & 0x1FFFFFFull) | (2u << 30); // [56:32] | type=2
  v8i g1;
  g1[0] = (int)(2u << 16);                  // data_size=2 (4 bytes), mask=0
  g1[1] = (int)((nelem & 0xFFFFu) << 16);   // tensor_dim0[15:0] at bits[63:48]
  g1[2] = (int)((nelem >> 16) | (1u << 16));// tensor_dim0[31:16] | tensor_dim1=1
  g1[3] = (int)(nelem << 16);               // tensor_dim1_hi=0 | tile_dim0=nelem
  g1[4] = 1;                                // tile_dim1=1, tile_dim2=0
  g1[5] = (int)nelem;                       // tensor_dim0_stride lo
  g1[6] = 0;                                // stride0 hi | stride1 lo
  g1[7] = 0;                                // stride1 hi
  v4i z4 = {0, 0, 0, 0};
  v8i z8 = {0, 0, 0, 0, 0, 0, 0, 0};
  __builtin_amdgcn_tensor_load_to_lds(g0, g1, z4, z4, z8, 0);
}
#endif

// b128-per-lane init: zero the 100.7 MB output, epsilon-fill the 33.5 MB wsum.
__global__ void init_kernel(float4* __restrict__ out, float4* __restrict__ wsum) {
  const size_t OUTN4 = (size_t)NBATCH * 3 * VRES * VRES * VRES / 4;
  const size_t WN4   = (size_t)NBATCH * VRES * VRES * VRES / 4;
  size_t stride = (size_t)gridDim.x * blockDim.x;
  size_t i0 = (size_t)blockIdx.x * blockDim.x + threadIdx.x;
  const float4 z = make_float4(0.0f, 0.0f, 0.0f, 0.0f);
  const float4 e = make_float4(WEIGHT_EPS, WEIGHT_EPS, WEIGHT_EPS, WEIGHT_EPS);
  for (size_t i = i0; i < OUTN4; i += stride) out[i] = z;
  for (size_t i = i0; i < WN4; i += stride) wsum[i] = e;
}

__global__ void splat_kernel(const float* __restrict__ verts,   // (B, 8000, 3)
                             const float* __restrict__ codes,   // (B, 6890, 3)
                             float* __restrict__ out,           // (B, 3, 128^3)
                             float* __restrict__ wsum) {        // (B, 128^3)
  __shared__ float s_pos[CHUNK * 3];
  __shared__ float s_code[CHUNK * 3];
  __shared__ float s_w[CHUNK * 21];   // 3 axes x 7 offsets per vertex
  __shared__ int   s_base[CHUNK * 3];

  const int chunks_per_b = (NSURF + CHUNK - 1) / CHUNK;
  const int b  = blockIdx.x / chunks_per_b;
  const int c0 = (blockIdx.x % chunks_per_b) * CHUNK;
  const int nv = min(CHUNK, NSURF - c0);

  const float* gpos  = verts + ((size_t)b * NVERT + c0) * 3;
  const float* gcode = codes + ((size_t)b * NSURF + c0) * 3;

#ifdef USE_TDM
  if (threadIdx.x == 0) {   // one wave issues both descriptors (EXEC ignored by TDM)
    tdm_load_1d((unsigned)(size_t)(void*)s_pos,  gpos,  (unsigned)(nv * 3));
    tdm_load_1d((unsigned)(size_t)(void*)s_code, gcode, (unsigned)(nv * 3));
#if __has_builtin(__builtin_amdgcn_s_wait_tensorcnt)
    __builtin_amdgcn_s_wait_tensorcnt(0);
#else
    asm volatile("s_wait_tensorcnt 0x0" ::: "memory");
#endif
  }
#else
  for (int t = threadIdx.x; t < nv * 3; t += blockDim.x) {
    s_pos[t]  = gpos[t];
    s_code[t] = gcode[t];
  }
#endif
  __syncthreads();

  // Separable per-axis Gaussian weights: axis a uses reversed coord p[a]=v[2-a].
  for (int t = threadIdx.x; t < nv * 21; t += blockDim.x) {
    int vi = t / 21;
    int r  = t - vi * 21;
    int a  = r / 7;
    int o  = r - a * 7;
    float pa   = s_pos[vi * 3 + (2 - a)];
    int  base  = (int)floorf((pa + 1.0f) * (VRES * 0.5f));
    int  idx   = base - RADIUS + o;
    float c    = (2.0f * (float)idx + 1.0f) * (1.0f / VRES) - 1.0f;
    float d    = c - pa;
    float w    = (idx >= 0 && idx < VRES) ? __expf(-INV_2SIG2 * d * d) : 0.0f;
    s_w[vi * 21 + a * 7 + o] = w;
    if (o == 0) s_base[vi * 3 + a] = base - RADIUS;
  }
  __syncthreads();

  const size_t vol = (size_t)VRES * VRES * VRES;
  float* outb = out  + (size_t)b * 3 * vol;
  float* wsb  = wsum + (size_t)b * vol;

  for (int t = threadIdx.x; t < nv * NCELLS; t += blockDim.x) {
    int vi   = t / NCELLS;
    int cell = t - vi * NCELLS;
    int oz   = cell / 49;
    int rem  = cell - oz * 49;
    int oy   = rem / 7;
    int ox   = rem - oy * 7;
    float w = s_w[vi * 21 + oz] * s_w[vi * 21 + 7 + oy] * s_w[vi * 21 + 14 + ox];
    if (w != 0.0f) {   // zero-weight (clipped) cells contribute exactly nothing
      int iz = s_base[vi * 3 + 0] + oz;
      int iy = s_base[vi * 3 + 1] + oy;
      int ix = s_base[vi * 3 + 2] + ox;
      size_t flat = ((size_t)iz * VRES + iy) * VRES + ix;
      atomicAdd(&outb[flat],           w * s_code[vi * 3 + 0]);
      atomicAdd(&outb[vol + flat],     w * s_code[vi * 3 + 1]);
      atomicAdd(&outb[2 * vol + flat], w * s_code[vi * 3 + 2]);
      atomicAdd(&wsb[flat], w);
    }
  }
}

// b128-per-lane normalize: out[b][ch][v] /= wsum[b][v], IEEE scalar divides.
__global__ void norm_kernel(float* __restrict__ out, const float* __restrict__ wsum) {
  const size_t vol  = (size_t)VRES * VRES * VRES;   // multiple of 4
  const size_t WN4  = (size_t)NBATCH * vol / 4;
  size_t stride = (size_t)gridDim.x * blockDim.x;
  for (size_t i4 = (size_t)blockIdx.x * blockDim.x + threadIdx.x; i4 < WN4;
       i4 += stride) {
    size_t i = i4 * 4;
    size_t b = i / vol;
    size_t v = i - b * vol;                 // 4-aligned, stays within one batch
    float4 w = *(const float4*)(wsum + i);
    float* o = out + b * 3 * vol + v;
    for (int ch = 0; ch < 3; ++ch) {
      float4 c = *(float4*)(o + (size_t)ch * vol);
      c.x = c.x / w.x;
      c.y = c.y / w.y;
      c.z = c.z / w.z;
      c.w = c.w / w.w;
      *(float4*)(o + (size_t)ch * vol) = c;
    }
  }
}

extern "C" void kernel_launch(void* const* d_in, const int* in_sizes, int n_in,
                              void* d_out, int out_size, void* d_ws, size_t ws_size,
                              hipStream_t stream) {
  const float* verts = (const float*)d_in[0];   // smpl_vertices   (4, 8000, 3) f32
  const float* codes = (const float*)d_in[1];   // smpl_vertex_code(4, 6890, 3) f32
  // d_in[2..4] (face codes / indices / tet indices) are dead in the reference.
  float* out  = (float*)d_out;                  // (4, 3, 128, 128, 128) f32
  float* wsum = (float*)d_ws;                   // 4 * 128^3 f32 = 33.5 MB scratch

  init_kernel<<<2048, 256, 0, stream>>>((float4*)out, (float4*)wsum);

  const int chunks_per_b = (NSURF + CHUNK - 1) / CHUNK;   // 216
  splat_kernel<<<NBATCH * chunks_per_b, 256, 0, stream>>>(verts, codes, out, wsum);

  norm_kernel<<<4096, 256, 0, stream>>>(out, wsum);
}